// EfficientGlobalPointer_7181185318931
// MI455X (gfx1250) — compile-verified
//
#include <hip/hip_runtime.h>

typedef __attribute__((ext_vector_type(2))) float v2f;
typedef __attribute__((ext_vector_type(8))) float v8f;

#define HIDDEN 64
#define NHEAD  16
#define SEQ    1024
#define DIN    768
#define NEGC   1000000000000.0f
// ln(10000)/32
#define LOG1E4_OVER_32 0.28782313662425576f

// ---------------- Stage 1: h = X@W1+b1, RoPE -> qw/kw, bias -> biasE/biasO ----
__global__ __launch_bounds__(128) void egp_stage1(
    const float* __restrict__ X,  const float* __restrict__ W1,
    const float* __restrict__ b1, const float* __restrict__ W2,
    const float* __restrict__ b2,
    float* __restrict__ qw, float* __restrict__ kw,
    float* __restrict__ biasE, float* __restrict__ biasO)
{
    __shared__ float sx[8][DIN];   // 8 input rows
    __shared__ float sh[8][128];   // 8 hidden rows
    const int  tid  = threadIdx.x;
    const long row0 = (long)blockIdx.x * 8;

    for (int i = tid; i < 8 * DIN; i += 128) {
        int r = i / DIN, c = i % DIN;
        sx[r][c] = X[(row0 + r) * DIN + c];
    }
    __syncthreads();

    // thread tid owns hidden column tid for all 8 rows
    float acc[8];
    #pragma unroll
    for (int r = 0; r < 8; ++r) acc[r] = b1[tid];
    for (int k = 0; k < DIN; ++k) {
        float w = W1[k * 128 + tid];          // coalesced
        #pragma unroll
        for (int r = 0; r < 8; ++r) acc[r] = fmaf(sx[r][k], w, acc[r]);
    }
    #pragma unroll
    for (int r = 0; r < 8; ++r) sh[r][tid] = acc[r];
    __syncthreads();

    // RoPE: qw[j]=h[2j], kw[j]=h[2j+1]; cos/sin index = j mod 32 (jnp.tile semantics)
    for (int idx = tid; idx < 8 * HIDDEN; idx += 128) {
        int  r   = idx / HIDDEN, j = idx % HIDDEN;
        long row = row0 + r;
        int  pos = (int)(row & (SEQ - 1));
        int  i32 = j & 31;
        float inv = __expf(-(float)i32 * LOG1E4_OVER_32);   // 10000^(-i/32)
        float ang = (float)pos * inv;
        float sv, cv;
        __sincosf(ang, &sv, &cv);
        float q  = sh[r][2 * j];
        float kk = sh[r][2 * j + 1];
        // rotate_half_interleaved on the 64-length vector: pairs (2i,2i+1)
        float qrot = (j & 1) ? sh[r][2 * (j - 1)]     : -sh[r][2 * (j + 1)];
        float krot = (j & 1) ? sh[r][2 * (j - 1) + 1] : -sh[r][2 * (j + 1) + 1];
        qw[row * HIDDEN + j] = fmaf(q,  cv, qrot * sv);
        kw[row * HIDDEN + j] = fmaf(kk, cv, krot * sv);
    }

    // bias = (h@W2 + b2)/2 ; split even heads (applied along n) / odd heads (along m)
    for (int idx = tid; idx < 8 * 32; idx += 128) {
        int  r   = idx / 32, c = idx % 32;
        long row = row0 + r;
        float a = b2[c];
        for (int k = 0; k < 128; ++k) a = fmaf(sh[r][k], W2[k * 32 + c], a);
        a *= 0.5f;
        if (c & 1) biasO[row * NHEAD + (c >> 1)] = a;
        else       biasE[row * NHEAD + (c >> 1)] = a;
    }
}

// ---------------- Stage 2: one 16x16 logits tile per wave, fan out 16 heads ---
__global__ __launch_bounds__(32) void egp_stage2(
    const float* __restrict__ qw, const float* __restrict__ kw,
    const float* __restrict__ biasE, const float* __restrict__ biasO,
    const float* __restrict__ mask, float* __restrict__ out)
{
    const int lane = threadIdx.x;
    const int n0   = blockIdx.x * 16;
    const int m0   = blockIdx.y * 16;
    const int bb   = blockIdx.z;
    const int half = lane >> 4;    // K sub-pair selector for A/B fragments
    const int l16  = lane & 15;

    __shared__ float sE[16 * 16];  // [n][head]
    __shared__ float sO[16 * 16];  // [m][head]
    for (int i = lane; i < 256; i += 32) {
        sE[i] = biasE[((long)bb * SEQ + n0 + (i >> 4)) * NHEAD + (i & 15)];
        sO[i] = biasO[((long)bb * SEQ + m0 + (i >> 4)) * NHEAD + (i & 15)];
    }

    // A: qw rows m0..m0+15 ; B: kw rows n0..n0+15 (B[k][n] = kw[n][k])
    const float* qrow = qw + ((long)bb * SEQ + m0 + l16) * HIDDEN + half * 2;
    const float* krow = kw + ((long)bb * SEQ + n0 + l16) * HIDDEN + half * 2;

    v8f acc = {};
    #pragma unroll
    for (int k = 0; k < 16; ++k) {          // K = 64 in steps of 4
        v2f av = *(const v2f*)(qrow + k * 4);
        v2f bv = *(const v2f*)(krow + k * 4);
        acc = __builtin_amdgcn_wmma_f32_16x16x4_f32(
            /*neg_a=*/false, av, /*neg_b=*/false, bv,
            /*c_mod=*/(short)0, acc, /*reuse_a=*/false, /*reuse_b=*/false);
    }

    const int  ng   = n0 + l16;             // D layout: lane -> column n
    const float padv = mask[(long)bb * SEQ + ng];
    __syncthreads();

    float* obase = out + ((long)bb * NHEAD) * SEQ * SEQ;
    #pragma unroll
    for (int hh = 0; hh < NHEAD; ++hh) {
        float be = sE[l16 * 16 + hh];
        #pragma unroll
        for (int r = 0; r < 8; ++r) {
            int   m  = r + half * 8;        // D layout: vgpr r -> row m (+8 upper half)
            int   mg = m0 + m;
            float v  = fmaf(acc[r], 0.125f, be + sO[m * 16 + hh]);
            v = v * padv - (1.0f - padv) * NEGC;      // pad mask
            if (mg > ng) v -= NEGC;                   // strict-lower causal
            __builtin_nontemporal_store(v, obase + ((long)hh * SEQ + mg) * SEQ + ng);
        }
    }
}

extern "C" void kernel_launch(void* const* d_in, const int* in_sizes, int n_in,
                              void* d_out, int out_size, void* d_ws, size_t ws_size,
                              hipStream_t stream) {
    const float* X    = (const float*)d_in[0];  // (4,1024,768)
    const float* mask = (const float*)d_in[1];  // (4,1024)
    const float* W1   = (const float*)d_in[2];  // (768,128)
    const float* b1   = (const float*)d_in[3];  // (128)
    const float* W2   = (const float*)d_in[4];  // (128,32)
    const float* b2   = (const float*)d_in[5];  // (32)
    float* out = (float*)d_out;                 // (4,16,1024,1024)

    float* ws = (float*)d_ws;
    float* qwp = ws;                              // 4*1024*64
    float* kwp = qwp + 4L * SEQ * HIDDEN;         // 4*1024*64
    float* bE  = kwp + 4L * SEQ * HIDDEN;         // 4*1024*16
    float* bO  = bE  + 4L * SEQ * NHEAD;          // 4*1024*16

    egp_stage1<<<(4 * SEQ) / 8, 128, 0, stream>>>(X, W1, b1, W2, b2, qwp, kwp, bE, bO);
    egp_stage2<<<dim3(SEQ / 16, SEQ / 16, 4), 32, 0, stream>>>(qwp, kwp, bE, bO, mask, out);
}